// PLA7_14705968022388
// MI455X (gfx1250) — compile-verified
//
#include <hip/hip_runtime.h>

// ---------------------------------------------------------------------------
// Problem constants (from reference): B=8, NF=40, C=120, H=W=128, HW=16384
// ---------------------------------------------------------------------------
#define BB      8
#define NFC     40
#define CCH     120          // concat channels
#define HWN     16384        // 128*128
#define KSPLIT  16
#define KCHUNK  (HWN / KSPLIT)   // 1024

typedef float v2f __attribute__((ext_vector_type(2)));
typedef float v8f __attribute__((ext_vector_type(8)));

__device__ __forceinline__ v8f wmma_f32_16x16x4(v2f a, v2f b, v8f c) {
  // (neg_a, A, neg_b, B, c_mod, C, reuse_a, reuse_b)
  return __builtin_amdgcn_wmma_f32_16x16x4_f32(false, a, false, b, (short)0, c,
                                               false, false);
}

__device__ __forceinline__ v8f v8zero() {
  v8f z = {0.f, 0.f, 0.f, 0.f, 0.f, 0.f, 0.f, 0.f};
  return z;
}

// Interleaved B-slab addressing: element (k, n) of a 128-wide slab lives at
// (k>>1)*256 + 2n + (k&1), so the lane's K pair is one aligned 8-byte word.
__device__ __forceinline__ int bidx(int k, int n) {
  return ((k >> 1) << 8) + (n << 1) + (k & 1);
}

// ---------------------------------------------------------------------------
// K1: p_feats[b][m][n] = relu( sum_k W_p[m][k] * P[b][k][n] ),  P = cat(f,s,c)
//     Block: 128-wide N slab, all 120(->128 padded) M rows, 8 waves.
//     A (W_p) read in-loop as L2-hot global b64 (register preload spills).
// ---------------------------------------------------------------------------
__global__ __launch_bounds__(256) void k1_gemm_relu(
    const float* __restrict__ f, const float* __restrict__ s,
    const float* __restrict__ cin, const float* __restrict__ w_p,
    float* __restrict__ pf) {
  __shared__ float sP[CCH * 128];          // interleaved [k-pair][n][2], 60 KB
  const int tid = threadIdx.x;
  const int b  = blockIdx.y;
  const int n0 = blockIdx.x * 128;

  for (int idx = tid; idx < CCH * 128; idx += 256) {
    const int ch = idx >> 7, col = idx & 127;
    const float* src = (ch < NFC) ? f : (ch < 2 * NFC) ? s : cin;
    const int ch3 = (ch < NFC) ? ch : (ch < 2 * NFC) ? ch - NFC : ch - 2 * NFC;
    sP[bidx(ch, col)] = src[((size_t)b * NFC + ch3) * HWN + n0 + col];
  }

  const int wave = tid >> 5, lane = tid & 31;
  const int half = lane >> 4, lrow = lane & 15;
  const int m  = wave * 16 + lrow;
  const int mA = m < CCH ? m : CCH - 1;     // clamp padded A rows (not stored)
  const float* __restrict__ arow = w_p + mA * CCH + 2 * half;

  v8f acc[8];
#pragma unroll
  for (int t = 0; t < 8; ++t) acc[t] = v8zero();

  __syncthreads();

#pragma unroll 6
  for (int kk = 0; kk < 30; ++kk) {
    const v2f a = *(const v2f*)(arow + 4 * kk);          // global, L2/L0 hit
    const int ka = 4 * kk + 2 * half;                    // even pair index
    const float* bp = sP + ((ka >> 1) << 8);
#pragma unroll
    for (int t = 0; t < 8; ++t) {
      const v2f bf = *(const v2f*)(bp + ((t * 16 + lrow) << 1));
      acc[t] = wmma_f32_16x16x4(a, bf, acc[t]);
    }
  }

#pragma unroll
  for (int t = 0; t < 8; ++t) {
#pragma unroll
    for (int r = 0; r < 8; ++r) {
      const int mo = wave * 16 + r + 8 * half;
      if (mo < CCH) {
        const float v = acc[t][r];
        pf[((size_t)b * CCH + mo) * HWN + n0 + t * 16 + lrow] =
            v > 0.f ? v : 0.f;
      }
    }
  }
}

// ---------------------------------------------------------------------------
// K2: attn partials. attn[i][j] = sum_k X[i][k] * flat[k*120 + j]
//     Block (ks,b): K range [ks*1024, +1024), full 128x128 padded output,
//     32-wide K slabs staged through LDS; deterministic split-K buffer.
// ---------------------------------------------------------------------------
__global__ __launch_bounds__(256) void k2_attn_partial(
    const float* __restrict__ pf, float* __restrict__ part) {
  __shared__ float sX[128 * 32];            // [m][k] row-major, pad rows zeroed
  __shared__ float sY[32 * 128];            // interleaved [k-pair][n][2]
  const int tid = threadIdx.x;
  const int b  = blockIdx.y;
  const int ks = blockIdx.x;
  const float* __restrict__ Xb = pf + (size_t)b * CCH * HWN;

  const int wave = tid >> 5, lane = tid & 31;
  const int half = lane >> 4, lrow = lane & 15;
  const int arow = (wave * 16 + lrow) * 32; // lane's A row base in sX

  v8f acc[8];
#pragma unroll
  for (int t = 0; t < 8; ++t) acc[t] = v8zero();

  const int kend = ks * KCHUNK + KCHUNK;
  for (int kb = ks * KCHUNK; kb < kend; kb += 32) {
    for (int idx = tid; idx < 128 * 32; idx += 256) {
      const int mm = idx >> 5, kk = idx & 31;
      sX[idx] = (mm < CCH) ? Xb[(size_t)mm * HWN + kb + kk] : 0.f;
    }
    for (int idx = tid; idx < 32 * 128; idx += 256) {
      const int kk = idx >> 7, nn = idx & 127;
      sY[bidx(kk, nn)] = (nn < CCH) ? Xb[(size_t)(kb + kk) * CCH + nn] : 0.f;
    }
    __syncthreads();
#pragma unroll
    for (int k0 = 0; k0 < 32; k0 += 4) {
      const int ka = k0 + 2 * half;
      const v2f a = *(const v2f*)(sX + arow + ka);
      const float* bp = sY + ((ka >> 1) << 8);
#pragma unroll
      for (int t = 0; t < 8; ++t) {
        const v2f bf = *(const v2f*)(bp + ((t * 16 + lrow) << 1));
        acc[t] = wmma_f32_16x16x4(a, bf, acc[t]);
      }
    }
    __syncthreads();
  }

  float* __restrict__ Pp = part + ((size_t)b * KSPLIT + ks) * 128 * 128;
#pragma unroll
  for (int t = 0; t < 8; ++t) {
#pragma unroll
    for (int r = 0; r < 8; ++r) {
      Pp[(wave * 16 + r + 8 * half) * 128 + t * 16 + lrow] = acc[t][r];
    }
  }
}

// ---------------------------------------------------------------------------
// K3: reduce split-K partials, row softmax, then M = scale1 * (W_c3 @ attnS).
//     One block per batch.
// ---------------------------------------------------------------------------
__global__ __launch_bounds__(256) void k3_softmax_M(
    const float* __restrict__ part, const float* __restrict__ w_c3,
    const float* __restrict__ scale1, float* __restrict__ Mmat) {
  __shared__ float sA[CCH * 128];           // attn rows, stride 128
  const int tid = threadIdx.x;
  const int b = blockIdx.x;

  for (int idx = tid; idx < CCH * 128; idx += 256) {
    const int mm = idx >> 7, nn = idx & 127;
    float sum = 0.f;
    if (nn < CCH) {
      for (int ks = 0; ks < KSPLIT; ++ks)
        sum += part[(((size_t)b * KSPLIT + ks) * 128 + mm) * 128 + nn];
    }
    sA[idx] = sum;
  }
  __syncthreads();

  const int wave = tid >> 5, lane = tid & 31;
  for (int row = wave; row < CCH; row += 8) {
    float* r = sA + row * 128;
    float mx = -3.4e38f;
    for (int j = lane; j < CCH; j += 32) mx = fmaxf(mx, r[j]);
#pragma unroll
    for (int off = 16; off; off >>= 1) mx = fmaxf(mx, __shfl_xor(mx, off, 32));
    float sm = 0.f;
    for (int j = lane; j < CCH; j += 32) {
      const float e = __expf(r[j] - mx);
      r[j] = e;
      sm += e;
    }
#pragma unroll
    for (int off = 16; off; off >>= 1) sm += __shfl_xor(sm, off, 32);
    const float inv = 1.f / sm;
    for (int j = lane; j < CCH; j += 32) r[j] *= inv;
  }
  __syncthreads();

  const float sc = scale1[0];
  for (int idx = tid; idx < NFC * CCH; idx += 256) {
    const int o = idx / CCH, j = idx % CCH;
    float sum = 0.f;
    for (int c = 0; c < CCH; ++c) sum += w_c3[o * CCH + c] * sA[c * 128 + j];
    Mmat[((size_t)b * NFC + o) * CCH + j] = sc * sum;
  }
}

// ---------------------------------------------------------------------------
// K4: out = M @ X + W_c3 @ P + f.  Two accumulation passes over one LDS slab.
//     M dim 40 -> 3 tiles (padded to 48 via clamped A rows / masked stores).
// ---------------------------------------------------------------------------
__global__ __launch_bounds__(256) void k4_out(
    const float* __restrict__ f, const float* __restrict__ s,
    const float* __restrict__ cin, const float* __restrict__ w_c3,
    const float* __restrict__ Mmat, const float* __restrict__ pf,
    float* __restrict__ out) {
  __shared__ float sB[CCH * 128];           // interleaved slab (X, then P)
  const int tid = threadIdx.x;
  const int b  = blockIdx.y;
  const int n0 = blockIdx.x * 128;
  const int wave = tid >> 5, lane = tid & 31;
  const int half = lane >> 4, lrow = lane & 15;
  const int ncol = wave * 16 + lrow;        // wave owns one 16-wide N subtile

  v8f acc[3];
#pragma unroll
  for (int t = 0; t < 3; ++t) acc[t] = v8zero();

  // ---- pass 1: acc += M @ X-slab ----
  for (int idx = tid; idx < CCH * 128; idx += 256) {
    const int ch = idx >> 7, col = idx & 127;
    sB[bidx(ch, col)] = pf[((size_t)b * CCH + ch) * HWN + n0 + col];
  }
  // hint the P slab we'll need next (global_prefetch_b8)
  __builtin_prefetch(f + ((size_t)b * NFC + (tid >> 7)) * HWN + n0 + (tid & 127),
                     0, 0);
  __syncthreads();

  const float* __restrict__ A1 = Mmat + (size_t)b * NFC * CCH;
  for (int k0 = 0; k0 < CCH; k0 += 4) {
    const int ka = k0 + 2 * half;
    const v2f bf = *(const v2f*)(sB + ((ka >> 1) << 8) + (ncol << 1));
#pragma unroll
    for (int tm = 0; tm < 3; ++tm) {
      const int mr = tm * 16 + lrow;
      const int mA = mr < NFC ? mr : NFC - 1;
      const v2f a = *(const v2f*)(A1 + mA * CCH + ka);
      acc[tm] = wmma_f32_16x16x4(a, bf, acc[tm]);
    }
  }
  __syncthreads();

  // ---- pass 2: acc += W_c3 @ P-slab ----
  for (int idx = tid; idx < CCH * 128; idx += 256) {
    const int ch = idx >> 7, col = idx & 127;
    const float* src = (ch < NFC) ? f : (ch < 2 * NFC) ? s : cin;
    const int ch3 = (ch < NFC) ? ch : (ch < 2 * NFC) ? ch - NFC : ch - 2 * NFC;
    sB[bidx(ch, col)] = src[((size_t)b * NFC + ch3) * HWN + n0 + col];
  }
  __syncthreads();

  for (int k0 = 0; k0 < CCH; k0 += 4) {
    const int ka = k0 + 2 * half;
    const v2f bf = *(const v2f*)(sB + ((ka >> 1) << 8) + (ncol << 1));
#pragma unroll
    for (int tm = 0; tm < 3; ++tm) {
      const int mr = tm * 16 + lrow;
      const int mA = mr < NFC ? mr : NFC - 1;
      const v2f a = *(const v2f*)(w_c3 + mA * CCH + ka);
      acc[tm] = wmma_f32_16x16x4(a, bf, acc[tm]);
    }
  }

  // ---- epilogue: + f residual, masked store (rows < 40) ----
#pragma unroll
  for (int tm = 0; tm < 3; ++tm) {
#pragma unroll
    for (int r = 0; r < 8; ++r) {
      const int mo = tm * 16 + r + 8 * half;
      if (mo < NFC) {
        const size_t gi = ((size_t)b * NFC + mo) * HWN + n0 + ncol;
        out[gi] = acc[tm][r] + f[gi];
      }
    }
  }
}

// ---------------------------------------------------------------------------
// Launch
// ---------------------------------------------------------------------------
extern "C" void kernel_launch(void* const* d_in, const int* in_sizes, int n_in,
                              void* d_out, int out_size, void* d_ws,
                              size_t ws_size, hipStream_t stream) {
  const float* f      = (const float*)d_in[0];
  const float* s      = (const float*)d_in[1];
  const float* c      = (const float*)d_in[2];
  const float* w_p    = (const float*)d_in[3];
  const float* w_c3   = (const float*)d_in[4];
  const float* scale1 = (const float*)d_in[5];
  float* out = (float*)d_out;

  float* ws   = (float*)d_ws;
  float* pf   = ws;                                        // B*120*16384
  float* part = pf + (size_t)BB * CCH * HWN;               // B*16*128*128
  float* Mmat = part + (size_t)BB * KSPLIT * 128 * 128;    // B*40*120

  dim3 blk(256);
  k1_gemm_relu<<<dim3(HWN / 128, BB), blk, 0, stream>>>(f, s, c, w_p, pf);
  k2_attn_partial<<<dim3(KSPLIT, BB), blk, 0, stream>>>(pf, part);
  k3_softmax_M<<<dim3(BB), blk, 0, stream>>>(part, w_c3, scale1, Mmat);
  k4_out<<<dim3(HWN / 128, BB), blk, 0, stream>>>(f, s, c, w_c3, Mmat, pf, out);
}